// AcousticRadianceTransfer_PatchDirectionFactorized_89893665505842
// MI455X (gfx1250) — compile-verified
//
#include <hip/hip_runtime.h>
#include <math.h>

// ---------------------------------------------------------------------------
// Acoustic radiance transfer on MI455X (gfx1250).
// Dense per-patch 128x128 real K times complex radiance is done with fp32 WMMA
// (V_WMMA_F32_16X16X4_F32): precision-safe, and the workload is bandwidth
// bound (~3 GB traffic vs 17 GFLOP => HBM/L2 roofline), so fp32 matrix ops
// cost nothing.  Each wave register-blocks a 32x32 output tile (2 d-tiles x
// 2 f-tiles, Re+Im) => 8 WMMAs per 12 global loads and 2x less redundant L2
// traffic on both the K and radiance operands.
// Frequency stride padded to 544 (34 WMMA n-tiles -> 17 tile pairs) so the
// WMMA path needs no guards and padding columns stay exactly zero everywhere.
// ---------------------------------------------------------------------------

typedef float v2f __attribute__((ext_vector_type(2)));
typedef float v8f __attribute__((ext_vector_type(8)));

constexpr int NP  = 128;      // patches
constexpr int ND  = 128;      // directions per patch
constexpr int NR  = 16384;    // NP*ND radiance rows
constexpr int NE  = 131072;   // sparse edges
constexpr int NT  = 1024;     // time samples
constexpr int NF  = 513;      // rfft bins
constexpr int NFP = 544;      // padded freq stride (34 * 16)

constexpr float SRATE    = 16000.0f;
constexpr float CSND     = 343.0f;
constexpr float LOGG     = -6.907755278982137f;    // ln(0.001)
constexpr float AIRA     = 0.001f;
constexpr float TAU_COEF = SRATE / CSND;           // dist -> delay (samples)
constexpr float DEC_COEF = LOGG / CSND - AIRA;     // exp coefficient on dist
constexpr float PI_F     = 3.14159265358979323846f;
constexpr float ANG_COEF = -2.0f * PI_F / (float)NT;

// ---------------- per-row geometric precompute ----------------
__global__ __launch_bounds__(256) void k_rowinit(
    const float* __restrict__ src, const float* __restrict__ rec,
    const float* __restrict__ avgd, const float* __restrict__ rpos,
    float* __restrict__ g_src, float* __restrict__ tau_src, float* __restrict__ dec_src,
    float* __restrict__ w_rec, float* __restrict__ tau_rec, float* __restrict__ dec_rec,
    float* __restrict__ tau_av, float* __restrict__ dec_av)
{
    int i = blockIdx.x * 256 + threadIdx.x;
    if (i >= NR) return;
    float px = rpos[i*3+0], py = rpos[i*3+1], pz = rpos[i*3+2];
    float dx = px - src[0], dy = py - src[1], dz = pz - src[2];
    float ds = sqrtf(dx*dx + dy*dy + dz*dz);
    g_src[i]   = 1.0f / (ds*ds + 0.001f);
    tau_src[i] = ds * TAU_COEF;
    dec_src[i] = expf(DEC_COEF * ds);
    dx = px - rec[0]; dy = py - rec[1]; dz = pz - rec[2];
    float dr = sqrtf(dx*dx + dy*dy + dz*dz);
    w_rec[i]   = 1.0f / (dr*dr + 0.001f);
    tau_rec[i] = dr * TAU_COEF;
    dec_rec[i] = expf(DEC_COEF * dr);
    float da = avgd[i];
    tau_av[i] = da * TAU_COEF;
    dec_av[i] = expf(DEC_COEF * da);
}

// ---------------- build Kt[p][e][d] = K[p][d][e] (transposed for A-frag loads)
__global__ __launch_bounds__(256) void k_buildKt(
    const float* __restrict__ absorb, const float* __restrict__ scat,
    const int* __restrict__ obj, const float* __restrict__ basis,
    float* __restrict__ Kt)
{
    int idx = blockIdx.x * 256 + threadIdx.x;
    if (idx >= NP * ND * ND) return;
    int p = idx >> 14;
    int d = (idx >> 7) & 127;
    int e = idx & 127;
    int o = obj[p];
    float refl = 1.0f - absorb[o];
    float sc   = scat[o];
    float c0 = refl * sc;
    float c1 = refl * (1.0f - sc);
    float b0 = basis[d * ND + e];
    float b1 = basis[ND * ND + d * ND + e];
    Kt[(size_t)p * (ND * ND) + (size_t)e * ND + d] = c0 * b0 + c1 * b1;
}

// ---------------- CSR build: count -> scan -> fill ----------------
__global__ __launch_bounds__(256) void k_count(const int* __restrict__ grow,
                                               int* __restrict__ counts)
{
    int e = blockIdx.x * 256 + threadIdx.x;
    if (e < NE) atomicAdd(&counts[grow[e]], 1);
}

__global__ __launch_bounds__(1024) void k_scan(const int* __restrict__ counts,
                                               int* __restrict__ row_ptr,
                                               int* __restrict__ cursor)
{
    __shared__ int sh[1024];
    int t = threadIdx.x;
    int local[16];
    int s = 0;
    for (int j = 0; j < 16; ++j) { local[j] = s; s += counts[t * 16 + j]; }
    sh[t] = s;
    __syncthreads();
    for (int off = 1; off < 1024; off <<= 1) {
        int v = 0;
        if (t >= off) v = sh[t - off];
        __syncthreads();
        if (t >= off) sh[t] += v;
        __syncthreads();
    }
    int base = (t > 0) ? sh[t - 1] : 0;
    for (int j = 0; j < 16; ++j) {
        int val = base + local[j];
        row_ptr[t * 16 + j] = val;
        cursor[t * 16 + j]  = val;
    }
    if (t == 1023) row_ptr[NR] = sh[1023];
}

__global__ __launch_bounds__(256) void k_fill(
    const int* __restrict__ grow, const int* __restrict__ gcol,
    const float* __restrict__ gval, int* __restrict__ cursor,
    int* __restrict__ cols, float* __restrict__ vals)
{
    int e = blockIdx.x * 256 + threadIdx.x;
    if (e >= NE) return;
    int r = grow[e];
    int pos = atomicAdd(&cursor[r], 1);
    cols[pos] = gcol[e];
    vals[pos] = gval[e];
}

// ---------------- initial radiance: rad = g_src * path_factor(d_src) ---------
__global__ __launch_bounds__(256) void k_rad(
    const float* __restrict__ g_src, const float* __restrict__ tau_src,
    const float* __restrict__ dec_src,
    float* __restrict__ re, float* __restrict__ im)
{
    size_t idx = (size_t)blockIdx.x * 256 + threadIdx.x;
    if (idx >= (size_t)NR * NFP) return;
    int i = (int)(idx / NFP);
    int f = (int)(idx % NFP);
    float vr = 0.f, vi = 0.f;
    if (f < NF) {
        float g = g_src[i] * dec_src[i];
        float ang = ANG_COEF * tau_src[i] * (float)f;
        float s, c;
        sincosf(ang, &s, &c);
        vr = g * c; vi = g * s;
    }
    re[idx] = vr; im[idx] = vi;
}

// ---------------- dense bounce GEMM via fp32 WMMA, fused with prop ----------
// One wave per (patch, 32-row d-tile pair, 32-col f-tile pair): 8 accumulator
// tiles (Re/Im x 2x2), 8 x v_wmma_f32_16x16x4_f32 per K-step of 4.
// A (16x4 f32): lanes 0-15 hold {K=0,K=1}, lanes 16-31 hold {K=2,K=3}.
// B (4x16 f32): lanes 0-15 = rows K=0/1, lanes 16-31 = rows K=2/3.
// D (16x16 f32): VGPR v -> row v (lanes<16) / row 8+v (lanes>=16).
__global__ __launch_bounds__(256) void k_gemm(
    const float* __restrict__ Kt,
    const float* __restrict__ inRe, const float* __restrict__ inIm,
    const float* __restrict__ tau_av, const float* __restrict__ dec_av,
    float* __restrict__ outRe, float* __restrict__ outIm)
{
    int tid  = threadIdx.x;
    int lane = tid & 31;
    int wave = tid >> 5;
    int fw   = blockIdx.x * 8 + wave;         // 128*4*17 = 8704 waves total
    int p    = fw / (4 * 17);
    int rem  = fw % (4 * 17);
    int dtp  = rem / 17;                      // d-tile pair: 0..3
    int ftp  = rem % 17;                      // f-tile pair: 0..16
    int d0 = dtp * 32, f0 = ftp * 32;
    int half = lane >> 4;
    int l    = lane & 15;

    const float* ktp = Kt + (size_t)p * (ND * ND);
    v8f aR[2][2], aI[2][2];
#pragma unroll
    for (int ti = 0; ti < 2; ++ti)
#pragma unroll
        for (int tj = 0; tj < 2; ++tj) {
            aR[ti][tj] = (v8f){0.f,0.f,0.f,0.f,0.f,0.f,0.f,0.f};
            aI[ti][tj] = (v8f){0.f,0.f,0.f,0.f,0.f,0.f,0.f,0.f};
        }

    for (int ke = 0; ke < ND; ke += 4) {
        int e0 = ke + half * 2;
        v2f a0, a1;
        a0.x = ktp[(size_t)(e0 + 0) * ND + d0 + l];
        a0.y = ktp[(size_t)(e0 + 1) * ND + d0 + l];
        a1.x = ktp[(size_t)(e0 + 0) * ND + d0 + 16 + l];
        a1.y = ktp[(size_t)(e0 + 1) * ND + d0 + 16 + l];
        size_t rowA = ((size_t)p * ND + e0) * NFP + f0 + l;
        v2f br0, bi0, br1, bi1;
        br0.x = inRe[rowA];      br0.y = inRe[rowA + NFP];
        bi0.x = inIm[rowA];      bi0.y = inIm[rowA + NFP];
        br1.x = inRe[rowA + 16]; br1.y = inRe[rowA + NFP + 16];
        bi1.x = inIm[rowA + 16]; bi1.y = inIm[rowA + NFP + 16];
        aR[0][0] = __builtin_amdgcn_wmma_f32_16x16x4_f32(false, a0, false, br0, (short)0, aR[0][0], false, false);
        aI[0][0] = __builtin_amdgcn_wmma_f32_16x16x4_f32(false, a0, false, bi0, (short)0, aI[0][0], false, false);
        aR[0][1] = __builtin_amdgcn_wmma_f32_16x16x4_f32(false, a0, false, br1, (short)0, aR[0][1], false, false);
        aI[0][1] = __builtin_amdgcn_wmma_f32_16x16x4_f32(false, a0, false, bi1, (short)0, aI[0][1], false, false);
        aR[1][0] = __builtin_amdgcn_wmma_f32_16x16x4_f32(false, a1, false, br0, (short)0, aR[1][0], false, false);
        aI[1][0] = __builtin_amdgcn_wmma_f32_16x16x4_f32(false, a1, false, bi0, (short)0, aI[1][0], false, false);
        aR[1][1] = __builtin_amdgcn_wmma_f32_16x16x4_f32(false, a1, false, br1, (short)0, aR[1][1], false, false);
        aI[1][1] = __builtin_amdgcn_wmma_f32_16x16x4_f32(false, a1, false, bi1, (short)0, aI[1][1], false, false);
    }

    // fused multiply by prop = decay_avg[i] * cis(ANG_COEF * tau_avg[i] * f)
#pragma unroll
    for (int ti = 0; ti < 2; ++ti) {
#pragma unroll
        for (int v = 0; v < 8; ++v) {
            int drow = d0 + 16 * ti + (half ? 8 + v : v);
            int i = p * ND + drow;
            float w   = dec_av[i];
            float tau = tau_av[i];
#pragma unroll
            for (int tj = 0; tj < 2; ++tj) {
                int f = f0 + 16 * tj + l;
                float ang = ANG_COEF * tau * (float)f;
                float s, c;
                sincosf(ang, &s, &c);
                float re = aR[ti][tj][v], im = aI[ti][tj][v];
                outRe[(size_t)i * NFP + f] = w * (re * c - im * s);
                outIm[(size_t)i * NFP + f] = w * (re * s + im * c);
            }
        }
    }
}

// ---------------- sparse segment-sum: new_r[i] = sum_e val * r2[col] --------
__global__ __launch_bounds__(128) void k_gather(
    const int* __restrict__ row_ptr, const int* __restrict__ cols,
    const float* __restrict__ vals,
    const float* __restrict__ inRe, const float* __restrict__ inIm,
    float* __restrict__ outRe, float* __restrict__ outIm)
{
    int i = blockIdx.x;
    int f = blockIdx.y * 128 + threadIdx.x;
    if (f >= NFP) return;
    int s = row_ptr[i], e = row_ptr[i + 1];
    float ar = 0.f, ai = 0.f;
    for (int k = s; k < e; ++k) {
        int j   = cols[k];
        float v = vals[k];
        ar += v * inRe[(size_t)j * NFP + f];
        ai += v * inIm[(size_t)j * NFP + f];
    }
    outRe[(size_t)i * NFP + f] = ar;
    outIm[(size_t)i * NFP + f] = ai;
}

// ---------------- fused receiver-weighted reduction into echo_f -------------
__global__ __launch_bounds__(128) void k_echo(
    const float* __restrict__ re, const float* __restrict__ im,
    const float* __restrict__ w_rec, const float* __restrict__ tau_rec,
    const float* __restrict__ dec_rec, float* __restrict__ echo)
{
    int f = blockIdx.y * 128 + threadIdx.x;
    if (f >= NF) return;
    int i0 = blockIdx.x * 256;
    float ar = 0.f, ai = 0.f;
    for (int i = i0; i < i0 + 256; ++i) {
        float w = w_rec[i] * dec_rec[i];
        float ang = ANG_COEF * tau_rec[i] * (float)f;
        float s, c;
        sincosf(ang, &s, &c);
        float rr = re[(size_t)i * NFP + f];
        float ri = im[(size_t)i * NFP + f];
        ar += w * (c * rr - s * ri);
        ai += w * (s * rr + c * ri);
    }
    atomicAdd(&echo[f], ar);
    atomicAdd(&echo[NF + f], ai);
}

// ---------------- 1024-pt c2r inverse DFT + FSM window undo -----------------
__global__ __launch_bounds__(256) void k_irfft(const float* __restrict__ echo,
                                               float* __restrict__ out)
{
    int t = blockIdx.x * 256 + threadIdx.x;
    if (t >= NT) return;
    float acc = echo[0];                      // DC: imag ignored (pocketfft)
    float w0 = 2.0f * PI_F / (float)NT * (float)t;
    for (int k = 1; k < NF - 1; ++k) {
        float ang = w0 * (float)k;
        float s, c;
        sincosf(ang, &s, &c);
        acc += 2.0f * (echo[k] * c - echo[NF + k] * s);
    }
    acc += echo[NF - 1] * ((t & 1) ? -1.0f : 1.0f);   // Nyquist
    out[t] = (acc / (float)NT) * expf(-LOGG * (float)t / SRATE);
}

// ---------------------------------------------------------------------------
extern "C" void kernel_launch(void* const* d_in, const int* in_sizes, int n_in,
                              void* d_out, int out_size, void* d_ws, size_t ws_size,
                              hipStream_t stream)
{
    (void)in_sizes; (void)n_in; (void)out_size; (void)ws_size;
    const float* src   = (const float*)d_in[0];
    const float* rec   = (const float*)d_in[1];
    const float* absb  = (const float*)d_in[2];
    const float* scat  = (const float*)d_in[3];
    const float* gval  = (const float*)d_in[4];
    const float* basis = (const float*)d_in[5];
    const float* avgd  = (const float*)d_in[6];
    const float* rpos  = (const float*)d_in[7];
    const int*   grow  = (const int*)d_in[8];
    const int*   gcol  = (const int*)d_in[9];
    const int*   obj   = (const int*)d_in[10];
    float* out = (float*)d_out;

    // workspace carve (~153 MB): ping-pong complex planes + Kt + CSR + scalars
    const size_t RFP = (size_t)NR * NFP;
    float* w = (float*)d_ws;
    float* bufA_re = w; w += RFP;
    float* bufA_im = w; w += RFP;
    float* bufB_re = w; w += RFP;
    float* bufB_im = w; w += RFP;
    float* Kt      = w; w += (size_t)NP * ND * ND;
    float* g_src   = w; w += NR;
    float* tau_src = w; w += NR;
    float* dec_src = w; w += NR;
    float* w_rec   = w; w += NR;
    float* tau_rec = w; w += NR;
    float* dec_rec = w; w += NR;
    float* tau_av  = w; w += NR;
    float* dec_av  = w; w += NR;
    float* vals    = w; w += NE;
    float* echo    = w; w += 2 * NF;
    int* ip      = (int*)w;
    int* counts  = ip; ip += NR;
    int* row_ptr = ip; ip += NR + 1;
    int* cursor  = ip; ip += NR;
    int* cols    = ip; ip += NE;

    hipMemsetAsync(counts, 0, NR * sizeof(int), stream);
    hipMemsetAsync(echo, 0, 2 * NF * sizeof(float), stream);

    k_rowinit<<<NR / 256, 256, 0, stream>>>(src, rec, avgd, rpos,
        g_src, tau_src, dec_src, w_rec, tau_rec, dec_rec, tau_av, dec_av);
    k_buildKt<<<(NP * ND * ND) / 256, 256, 0, stream>>>(absb, scat, obj, basis, Kt);
    k_count<<<NE / 256, 256, 0, stream>>>(grow, counts);
    k_scan<<<1, 1024, 0, stream>>>(counts, row_ptr, cursor);
    k_fill<<<NE / 256, 256, 0, stream>>>(grow, gcol, gval, cursor, cols, vals);

    k_rad<<<(int)(RFP / 256), 256, 0, stream>>>(g_src, tau_src, dec_src,
                                                bufA_re, bufA_im);

    dim3 egrid(NR / 256, (NF + 127) / 128);
    dim3 ggrid(NR, (NFP + 127) / 128);
    k_echo<<<egrid, 128, 0, stream>>>(bufA_re, bufA_im, w_rec, tau_rec, dec_rec, echo);

    const int gemm_blocks = (NP * 4 * 17) / 8;   // 8704 waves / 8 per block
    for (int b = 0; b < 4; ++b) {
        k_gemm<<<gemm_blocks, 256, 0, stream>>>(Kt, bufA_re, bufA_im,
                                                tau_av, dec_av, bufB_re, bufB_im);
        k_gather<<<ggrid, 128, 0, stream>>>(row_ptr, cols, vals,
                                            bufB_re, bufB_im, bufA_re, bufA_im);
        k_echo<<<egrid, 128, 0, stream>>>(bufA_re, bufA_im, w_rec, tau_rec,
                                          dec_rec, echo);
    }

    k_irfft<<<NT / 256, 256, 0, stream>>>(echo, out);
}